// DiseaseDynamics_34024730919265
// MI455X (gfx1250) — compile-verified
//
#include <hip/hip_runtime.h>

// ---------------------------------------------------------------------------
// Dengue-style SEIR monthly scan for MI455X (gfx1250).
//
// Latency-bound problem: 36,000-step serial recurrence, ~30 KB total I/O
// (orders of magnitude under both the 23.3 TB/s HBM roof and the WMMA roof).
// One fused kernel, one workgroup:
//   Phase 1: wave 0 reduces sum(A_series) via V_WMMA_F32_16X16X4_F32
//            (fp32 tensor-core reduction: D = A x ones + C), branch-free
//            b64 loads for full tiles + cndmask-masked tail (EXEC all-1s).
//   Phase 2: all 256 threads compute the per-month force table into LDS.
//   Phase 3: lane 0 runs the exact serial day/month recurrence.
// ---------------------------------------------------------------------------

typedef __attribute__((ext_vector_type(2))) float v2f;
typedef __attribute__((ext_vector_type(8))) float v8f;

#define NHOST    14000000.0f
#define SIGMA_H  (1.0f / 5.5f)
#define GAMMA_C  (1.0f / 7.0f)
#define CAPV     (NHOST * 0.001f)      // max daily compartment increment
#define NMAX     2048                  // >= n_months (1200)

__global__ __launch_bounds__(256)
void dengue_scan_kernel(const float* __restrict__ A,      // [n] mosquito abundance
                        const float* __restrict__ W,      // [n,4] weather (col 0 = T)
                        const float* __restrict__ p_log_beta,
                        const float* __restrict__ p_log_import,
                        const float* __restrict__ p_log_amp,
                        const int*   __restrict__ p_dpm,
                        float* __restrict__ out,          // [n] monthly cases
                        int n)
{
    __shared__ float s_force[NMAX];
    __shared__ float s_sum;

    const int tid = threadIdx.x;

    // ---- Phase 1: sum(A) on wave 0 via fp32 WMMA (16x16x4, B = ones) ------
    if (tid < 32) {
        v8f c = {0.f, 0.f, 0.f, 0.f, 0.f, 0.f, 0.f, 0.f};
        v2f b; b.x = 1.0f; b.y = 1.0f;        // all-ones B => D rows = row-sums of A

        // Full 64-element tiles: unconditional 8-byte vector loads, zero
        // branches, zero exec manipulation inside the accumulation loop.
        const int nfull = n >> 6;             // n / 64
        for (int t = 0; t < nfull; ++t) {
            v2f a = *(const v2f*)(A + t * 64 + tid * 2);   // global_load_b64
            c = __builtin_amdgcn_wmma_f32_16x16x4_f32(
                    /*neg_a=*/false, a, /*neg_b=*/false, b,
                    /*c_mod=*/(short)0, c, /*reuse_a=*/false, /*reuse_b=*/false);
        }

        // Tail tile: clamp indices so loads are unconditional; masking is a
        // register-level v_cndmask, so EXEC remains all-1s at the WMMA.
        if (n & 63) {
            const int i0   = nfull * 64 + tid * 2;
            const int i1   = i0 + 1;
            const int last = n - 1;
            const float a0 = A[i0 < last ? i0 : last];
            const float a1 = A[i1 < last ? i1 : last];
            v2f a;
            a.x = (i0 < n) ? a0 : 0.0f;
            a.y = (i1 < n) ? a1 : 0.0f;
            c = __builtin_amdgcn_wmma_f32_16x16x4_f32(
                    false, a, false, b, (short)0, c, false, false);
        }

        // C/D layout: lanes 0-15 hold M=0..7 in VGPRs 0..7, lanes 16-31 hold
        // M=8..15. Every column of D equals the row-sum vector, so:
        //   total = (sum of 8 VGPRs in lane 0) + (sum of 8 VGPRs in lane 16)
        float s = c[0] + c[1] + c[2] + c[3] + c[4] + c[5] + c[6] + c[7];
        float tot = __shfl(s, 0, 32) + __shfl(s, 16, 32);
        if (tid == 0) s_sum = tot;
    }
    __syncthreads();

    // ---- Phase 2: per-month force-of-infection table (parallel, in LDS) ---
    const float beta      = fminf(fmaxf(expf(p_log_beta[0]), 1e-6f), 50.0f);
    const float amp       = expf(p_log_amp[0]);
    const float imp_daily = expf(p_log_import[0]) * (1.0f / 30.0f);
    const float A_mean    = s_sum / (float)n;
    const float inv_den   = 1.0f / (A_mean + 1.0f);

    for (int i = tid; i < n && i < NMAX; i += blockDim.x) {
        const float T    = W[i * 4];                              // temperature
        const float gate = (T > 14.0f && T < 35.0f) ? 1.0f : 0.0f;
        const float u    = (T - 27.0f) * (1.0f / 6.0f);
        const float bT   = 0.4f * expf(-u * u) * gate + 0.001f;   // Liu-Helmersson
        const float An   = A[i] * inv_den;
        s_force[i] = fminf(beta * bT * An / NHOST, 0.01f);
    }
    __syncthreads();

    // ---- Phase 3: serial 1200x30 recurrence (irreducible dep chain) -------
    if (tid == 0) {
        const int dpm = p_dpm[0];
        float Eh = 0.0f, Ih = 1.0f, Rh = 0.0f;
        for (int m = 0; m < n; ++m) {
            const float f_amp = s_force[m] * amp;   // hoisted: f * amp
            float mc = 0.0f;
            for (int d = 0; d < dpm; ++d) {
                const float Sh     = fmaxf(NHOST - Eh - Ih - Rh, 0.0f);
                const float Eh_pos = fmaxf(Eh, 0.0f);
                const float dEh    = f_amp * Sh + imp_daily - SIGMA_H * Eh;
                const float dIh    = SIGMA_H * Eh_pos - GAMMA_C * Ih;
                const float dRh    = GAMMA_C * fmaxf(Ih, 0.0f);
                mc += SIGMA_H * Eh_pos;                           // DT = 1
                Eh = fmaxf(Eh + fminf(fmaxf(dEh, -0.5f * Eh), CAPV), 0.0f);
                Ih = fmaxf(Ih + fminf(fmaxf(dIh, -0.5f * Ih), CAPV), 0.0f);
                Rh = fmaxf(Rh + dRh, 0.0f);
            }
            out[m] = mc;
        }
    }
}

extern "C" void kernel_launch(void* const* d_in, const int* in_sizes, int n_in,
                              void* d_out, int out_size, void* d_ws, size_t ws_size,
                              hipStream_t stream) {
    const float* A   = (const float*)d_in[0];   // A_series      [n]
    const float* W   = (const float*)d_in[1];   // weather_raw   [n,4]
    const float* lb  = (const float*)d_in[2];   // log_beta      [1]
    const float* li  = (const float*)d_in[3];   // log_import    [1]
    const float* la  = (const float*)d_in[4];   // log_amp       [1]
    const int*   dpm = (const int*)  d_in[5];   // days_per_month[1]
    float* out = (float*)d_out;
    const int n = in_sizes[0];

    // One workgroup (8 wave32 waves): parallel setup phases + serial scan.
    dengue_scan_kernel<<<1, 256, 0, stream>>>(A, W, lb, li, la, dpm, out, n);
    (void)d_ws; (void)ws_size; (void)n_in; (void)out_size;
}